// TSModel_34677565948107
// MI455X (gfx1250) — compile-verified
//
#include <hip/hip_runtime.h>

#define T_STEPS 2048
#define BROWS   16            // batch rows per block
#define HSTRIDE 66            // padded h1 row stride (floats) -> conflict-free banks
#define HSBUF   (16 * HSTRIDE)
#define XS_FLOATS (BROWS * T_STEPS)
#define SMEM_FLOATS (XS_FLOATS + 2 * HSBUF + 4 * HSTRIDE)
#define SMEM_BYTES  (SMEM_FLOATS * 4)

typedef float v2f __attribute__((ext_vector_type(2)));
typedef float v8f __attribute__((ext_vector_type(8)));

__device__ __forceinline__ float sig_fast(float x) {
  float e = __builtin_amdgcn_exp2f(-1.44269504f * x);   // 2^(-x*log2e) = e^-x
  return __builtin_amdgcn_rcpf(1.0f + e);
}
__device__ __forceinline__ float tanh_fast(float x) {
  float e = __builtin_amdgcn_exp2f(2.88539008f * x);    // e^(2x)
  return 1.0f - 2.0f * __builtin_amdgcn_rcpf(1.0f + e);
}

__global__ __launch_bounds__(288) void lstm2_kernel(
    const float* __restrict__ x,  const float* __restrict__ W1,
    const float* __restrict__ b1, const float* __restrict__ W2,
    const float* __restrict__ b2, float* __restrict__ y) {
  extern __shared__ float smem[];
  float* xs  = smem;                          // [16][2048] staged input
  float* hs  = smem + XS_FLOATS;              // 2 x [16][HSTRIDE] h1 double buffer
  float* w2t = smem + XS_FLOATS + 2 * HSBUF;  // [4][HSTRIDE] transposed W2 hidden part

  const int tid  = threadIdx.x;
  const int lane = tid & 31;
  const int wid  = tid >> 5;                  // 0..8 (9 waves)
  const int m0   = blockIdx.x * BROWS;

  // ---- stage x slice into LDS (float4, coalesced) ----
  for (int idx = tid; idx < BROWS * (T_STEPS / 4); idx += blockDim.x) {
    int m = idx >> 9, t4 = idx & 511;
    float4 v = reinterpret_cast<const float4*>(x + (size_t)(m0 + m) * T_STEPS)[t4];
    reinterpret_cast<float4*>(xs)[m * (T_STEPS / 4) + t4] = v;
  }
  // ---- zero h1 double buffer (h1(-1) = 0) ----
  for (int idx = tid; idx < 2 * HSBUF; idx += blockDim.x) hs[idx] = 0.0f;
  // ---- transpose W2 hidden rows into LDS: w2t[g][k] = W2[k][g] ----
  for (int idx = tid; idx < 4 * 64; idx += blockDim.x) {
    int g = idx >> 6, k = idx & 63;
    w2t[g * HSTRIDE + k] = W2[k * 4 + g];
  }
  __syncthreads();

  const int n  = lane & 15;
  const int hi = lane >> 4;

  // per-wave constant fragments
  v2f   bfrag[2][16];
  float b1v[2] = {0.f, 0.f}, wxv[2] = {0.f, 0.f};
  float b2v[2] = {0.f, 0.f}, w2h[2] = {0.f, 0.f};
  float c2r = 0.0f, h2r = 0.0f;
  float c1r[8];
#pragma unroll
  for (int r = 0; r < 8; ++r) c1r[r] = 0.0f;

  if (wid < 8) {
    // wave w owns hidden units u in [8w, 8w+8); tile0 = {i|j} cols, tile1 = {f|o} cols
    int u = 8 * wid + (n & 7);
    int col[2];
    col[0] = ((n < 8) ? 0 : 1) * 64 + u;   // gate i (n<8) / j (n>=8)
    col[1] = ((n < 8) ? 2 : 3) * 64 + u;   // gate f (n<8) / o (n>=8)
#pragma unroll
    for (int q = 0; q < 2; ++q) {
      b1v[q] = b1[col[q]];
      wxv[q] = W1[col[q]];                 // W1 row 0 = x weight
#pragma unroll
      for (int ks = 0; ks < 16; ++ks) {
        int k = 4 * ks + 2 * hi;           // B frag: lanes 0-15 hold K=k,k+1; 16-31 K=k+2,k+3
        v2f bb;
        bb.x = W1[(size_t)(1 + k) * 256 + col[q]];
        bb.y = W1[(size_t)(2 + k) * 256 + col[q]];
        bfrag[q][ks] = bb;
      }
    }
  } else {
    int gA = hi, gB = hi + 2;              // lanes<16: (i,f); lanes>=16: (j,o)
    b2v[0] = b2[gA];          b2v[1] = b2[gB];
    w2h[0] = W2[64 * 4 + gA]; w2h[1] = W2[64 * 4 + gB];
  }

  // ---- main recurrence: one barrier per step; layer2 pipelined one step behind ----
  for (int t = 0; t <= T_STEPS; ++t) {
    const int cur = t & 1;
    const float* hrd = hs + cur * HSBUF;         // h1(t-1)
    float*       hwr = hs + (cur ^ 1) * HSBUF;   // h1(t) destination

    if (wid < 8) {
      if (t < T_STEPS) {
        // A fragments: h1(t-1)[m][k], m = lane&15, K-pair selected by lane half
        v2f afrag[16];
#pragma unroll
        for (int ks = 0; ks < 16; ++ks)
          afrag[ks] = *reinterpret_cast<const v2f*>(hrd + n * HSTRIDE + 4 * ks + 2 * hi);
        float xt[8];
#pragma unroll
        for (int r = 0; r < 8; ++r) xt[r] = xs[(r + 8 * hi) * T_STEPS + t];

        v8f zt[2];
#pragma unroll
        for (int q = 0; q < 2; ++q) {
          v8f a0, a1;
#pragma unroll
          for (int r = 0; r < 8; ++r) { a0[r] = b1v[q] + xt[r] * wxv[q]; a1[r] = 0.0f; }
          // two independent 8-deep fp32 WMMA chains per tile (K=64 total), then merge
#pragma unroll
          for (int ks = 0; ks < 8; ++ks) {
            a0 = __builtin_amdgcn_wmma_f32_16x16x4_f32(
                false, afrag[2 * ks],     false, bfrag[q][2 * ks],     (short)0, a0, false, false);
            a1 = __builtin_amdgcn_wmma_f32_16x16x4_f32(
                false, afrag[2 * ks + 1], false, bfrag[q][2 * ks + 1], (short)0, a1, false, false);
          }
          zt[q] = a0 + a1;
        }

        // gate combine: i,f local; j,o live 8 lanes over -> shuffle, no LDS round trip
#pragma unroll
        for (int r = 0; r < 8; ++r) {
          float iv = zt[0][r];
          float jv = __shfl_xor(zt[0][r], 8);
          float fv = zt[1][r];
          float ov = __shfl_xor(zt[1][r], 8);
          float c  = sig_fast(fv + 1.0f) * c1r[r] + sig_fast(iv) * tanh_fast(jv);
          c1r[r]   = c;
          float hv = sig_fast(ov) * tanh_fast(c);
          if (n < 8) hwr[(r + 8 * hi) * HSTRIDE + 8 * wid + n] = hv;  // h1(t)[m][u]
        }
      }
    } else {
      // wave 8: layer2 for time t-1, overlapped with layer1 of time t
      if (t >= 1) {
        int m = lane & 15;
        float h2b = __shfl(h2r, m);               // broadcast h2[m] from lanes 0-15
        float aA = b2v[0] + h2b * w2h[0];         // lanes<16: i-dot ; lanes>=16: j-dot
        float aB = b2v[1] + h2b * w2h[1];         // lanes<16: f-dot ; lanes>=16: o-dot
        const float* hr = hrd + m * HSTRIDE;
        const float* wA = w2t + hi * HSTRIDE;
        const float* wB = w2t + (hi + 2) * HSTRIDE;
#pragma unroll
        for (int k = 0; k < 64; k += 2) {
          v2f hh = *reinterpret_cast<const v2f*>(hr + k);
          v2f wa = *reinterpret_cast<const v2f*>(wA + k);
          v2f wb = *reinterpret_cast<const v2f*>(wB + k);
          aA += hh.x * wa.x + hh.y * wa.y;
          aB += hh.x * wb.x + hh.y * wb.y;
        }
        float jv = __shfl_xor(aA, 16);
        float ov = __shfl_xor(aB, 16);
        float c  = sig_fast(aB + 1.0f) * c2r + sig_fast(aA) * tanh_fast(jv);
        c2r = c;
        h2r = sig_fast(ov) * tanh_fast(c);
        if (lane < 16) y[(size_t)(m0 + m) * T_STEPS + (t - 1)] = h2r;
      }
    }
    __syncthreads();
  }
}

extern "C" void kernel_launch(void* const* d_in, const int* in_sizes, int n_in,
                              void* d_out, int out_size, void* d_ws, size_t ws_size,
                              hipStream_t stream) {
  const float* x  = (const float*)d_in[0];   // [256, 2048, 1]
  const float* W1 = (const float*)d_in[1];   // [65, 256]
  const float* b1 = (const float*)d_in[2];   // [256]
  const float* W2 = (const float*)d_in[3];   // [65, 4]
  const float* b2 = (const float*)d_in[4];   // [4]
  float* y = (float*)d_out;                  // [256, 2048, 1]

  (void)in_sizes; (void)n_in; (void)out_size; (void)d_ws; (void)ws_size;

  (void)hipFuncSetAttribute(reinterpret_cast<const void*>(&lstm2_kernel),
                            hipFuncAttributeMaxDynamicSharedMemorySize,
                            (int)SMEM_BYTES);
  hipLaunchKernelGGL(lstm2_kernel, dim3(16), dim3(288), SMEM_BYTES, stream,
                     x, W1, b1, W2, b2, y);
}